// FloodGraph_23759759081724
// MI455X (gfx1250) — compile-verified
//
#include <hip/hip_runtime.h>

typedef float v2f __attribute__((ext_vector_type(2)));
typedef float v8f __attribute__((ext_vector_type(8)));

#define BATCH   2
#define NPTS    8192
#define KNN     30
#define MAXF    3.4028234663852886e38f

#define ROWS    16            // rows per block (one WMMA tile of rows)
#define WAVES   8             // wave32 waves per block
#define CHUNK   128           // columns per iteration = WAVES * 16
#define NCHUNK  (NPTS / CHUNK)
#define SCAN    8             // scanner threads per row (threads 0..127 scan)
#define PERSCAN (CHUNK / SCAN)
#define NLISTS  (ROWS * SCAN * KNN)

typedef unsigned long long u64;
typedef unsigned int u32;

// ---------------------------------------------------------------------------
// Kernel 1: pack node table P[b][n] = (x, y, z, w) with w = (C>0) ? |x|^2 : -1.
// |x|^2 >= 0, so the sign of w is a free mask bit. X is (B,N,1,3): the
// type-mean is the identity.
// ---------------------------------------------------------------------------
__global__ void flood_prep_kernel(const float* __restrict__ X,
                                  const long long* __restrict__ C,
                                  float4* __restrict__ P) {
  int i = blockIdx.x * blockDim.x + threadIdx.x;
  if (i >= BATCH * NPTS) return;
  float x = X[i * 3 + 0];
  float y = X[i * 3 + 1];
  float z = X[i * 3 + 2];
  float x2 = x * x + y * y + z * z;
  P[i] = make_float4(x, y, z, (C[i] > 0) ? x2 : -1.0f);
}

// ---------------------------------------------------------------------------
// Kernel 2: fused Gram (WMMA f32 16x16x4) + mask + top-30 + mask gather.
// One block = 16 rows of one batch; 8 wave32 waves stream 128 cols/iteration.
// Selection keys: (float_bits(d2) << 32) | col. d2 >= 0, so ascending u64
// order == lexicographic (value, index) order == JAX top_k tie-break.
// Hot path keeps only the cached worst key in VGPRs; the sorted 30-entry
// lists live in LDS (insertions are rare), so nothing spills to scratch.
// ---------------------------------------------------------------------------
__launch_bounds__(256, 1)
__global__ void flood_knn_kernel(const float4* __restrict__ P,
                                 float* __restrict__ outIdx,
                                 float* __restrict__ outMask) {
  __shared__ float sG[ROWS][CHUNK];        // Gram tile staging
  __shared__ float sX2[CHUNK];             // col |x|^2, or negative if masked
  __shared__ u64   sKey[ROWS][SCAN][KNN];  // per-scanner sorted key lists

  const int b    = blockIdx.y;
  const int i0   = blockIdx.x * ROWS;
  const int t    = threadIdx.x;
  const int wave = t >> 5;
  const int lane = t & 31;
  const int li   = lane & 15;   // row (A) / col (B) / col (D) index in tile
  const int kh   = lane >> 4;   // K half: 0 -> K0,K1 ; 1 -> K2,K3

  // Initialize LDS lists to the all-ones sentinel (larger than any valid key).
  {
    u64* flat = &sKey[0][0][0];
    for (int idx = t; idx < NLISTS; idx += 256) flat[idx] = ~0ull;
  }

  // A operand (row tile), identical for all waves. K=3 padded to 4 with 0.
  const float4 pr = P[(long long)b * NPTS + i0 + li];
  v2f a;
  a.x = kh ? pr.z : pr.x;
  a.y = kh ? 0.0f : pr.y;

  const int r = t & 15;    // scanner's row within tile
  const int s = t >> 4;    // scanner id (active when s < SCAN, i.e. t < 128)
  const float4 prow = P[(long long)b * NPTS + i0 + r];
  const float x2r = prow.w;
  const bool rowMasked = (prow.w < 0.0f);
  u64* const myList = &sKey[r][s & (SCAN - 1)][0];
  u64 worst = ~0ull;       // cached copy of myList[KNN-1]

#pragma unroll 1
  for (int ch = 0; ch < NCHUNK; ++ch) {
    // ---- phase 1: each wave computes one 16x16 Gram tile via WMMA ----
    const int jc = ch * CHUNK + wave * 16;
    const float4 pc = P[(long long)b * NPTS + jc + li];
    v2f bb;
    bb.x = kh ? pc.z : pc.x;
    bb.y = kh ? 0.0f : pc.y;
    v8f c = {};
    c = __builtin_amdgcn_wmma_f32_16x16x4_f32(
        /*neg_a=*/false, a, /*neg_b=*/false, bb,
        /*c_mod=*/(short)0, c, /*reuse_a=*/false, /*reuse_b=*/false);

    // D layout: lane holds column n=li; VGPR v holds row m = v + 8*kh.
#pragma unroll
    for (int v = 0; v < 8; ++v)
      sG[8 * kh + v][wave * 16 + li] = c[v];
    if (kh == 0) sX2[wave * 16 + li] = pc.w;   // sign carries the column mask
    __syncthreads();

    // ---- phase 2: 128 scanner threads update their LDS top-30 lists ----
    if (s < SCAN) {
#pragma unroll 1
      for (int q = 0; q < PERSCAN; ++q) {
        const int col = s * PERSCAN + q;
        const int j   = ch * CHUNK + col;
        const float g   = sG[r][col];
        const float x2c = sX2[col];
        float d2 = fmaxf(x2r + x2c - 2.0f * g, 0.0f);
        const float val = (rowMasked || x2c < 0.0f) ? MAXF : d2;
        const u64 key = ((u64)__float_as_uint(val) << 32) | (u32)j;
        if (key < worst) {                    // rare (~100x per scanner total)
          int k = KNN - 2;
#pragma unroll 1
          while (k >= 0) {                    // shift larger entries down
            const u64 e = myList[k];
            if (e <= key) break;
            myList[k + 1] = e;
            --k;
          }
          myList[k + 1] = key;
          worst = myList[KNN - 1];
        }
      }
    }
    __syncthreads();
  }

  // ---- final 8-way merge: one thread per row (lists already in LDS) ----
  if (t < ROWS) {
    int h[SCAN];
#pragma unroll
    for (int s2 = 0; s2 < SCAN; ++s2) h[s2] = 0;
    const float pwRow = P[(long long)b * NPTS + i0 + t].w;
    const float mrow = (pwRow < 0.0f) ? 0.0f : 1.0f;
    const long long ob = ((long long)b * NPTS + i0 + t) * KNN;
#pragma unroll 1
    for (int k = 0; k < KNN; ++k) {
      u64 best = ~0ull;
      int bests = 0;
#pragma unroll
      for (int s2 = 0; s2 < SCAN; ++s2) {
        const u64 v = (h[s2] < KNN) ? sKey[t][s2][h[s2]] : ~0ull;
        const bool lt = v < best;
        best  = lt ? v : best;
        bests = lt ? s2 : bests;
      }
      h[bests]++;
      const int besti = (int)(best & 0xffffffffu);
      const float mcol = (P[(long long)b * NPTS + besti].w < 0.0f) ? 0.0f : 1.0f;
      outIdx[ob + k]  = (float)besti;                  // exact for idx < 2^24
      outMask[ob + k] = mrow * mcol;
    }
  }
}

// ---------------------------------------------------------------------------
extern "C" void kernel_launch(void* const* d_in, const int* in_sizes, int n_in,
                              void* d_out, int out_size, void* d_ws, size_t ws_size,
                              hipStream_t stream) {
  (void)in_sizes; (void)n_in; (void)out_size; (void)ws_size;
  const float*     X = (const float*)d_in[0];       // (2, 8192, 1, 3) f32
  const long long* C = (const long long*)d_in[1];   // (2, 8192) i64

  float4* P = (float4*)d_ws;                        // 2*8192*16 B scratch

  flood_prep_kernel<<<(BATCH * NPTS + 255) / 256, 256, 0, stream>>>(X, C, P);

  float* outIdx  = (float*)d_out;                          // (2, 8192, 30)
  float* outMask = outIdx + (size_t)BATCH * NPTS * KNN;    // (2, 8192, 30)

  dim3 grid(NPTS / ROWS, BATCH);   // 512 x 2 blocks, 256 threads (8 wave32)
  flood_knn_kernel<<<grid, 256, 0, stream>>>(P, outIdx, outMask);
}